// L_G_87411174409129
// MI455X (gfx1250) — compile-verified
//
#include <hip/hip_runtime.h>
#include <math.h>

// ext-vector types (native clang vectors: safe for WMMA builtins & nontemporal loads)
typedef float v2f __attribute__((ext_vector_type(2)));
typedef float v8f __attribute__((ext_vector_type(8)));
typedef float f4  __attribute__((ext_vector_type(4)));

#define TPB 256

// Full-wave (32 lane) sum. Uses V_WMMA_F32_16X16X4_F32 as a 32->16 reducer:
//   A = ones(16x4), B: vgpr0 = per-lane partial, vgpr1 = 0
//   => D[m][n] = sum_k B[k][n] = acc[n] + acc[n+16]  (replicated over all m)
// then 4 xor-shuffles finish 16->1 (every lane ends with the wave total).
// Requires EXEC == all ones (callers guarantee no divergence at call site).
__device__ __forceinline__ float wave_sum32(float acc) {
  v2f a;  a[0] = 1.0f; a[1] = 1.0f;   // ones in every A slot (unused K rows hit zero B rows)
  v2f bm; bm[0] = acc; bm[1] = 0.0f;  // partials occupy 2 of 4 K-rows, rest zero
  v8f c = {};
  c = __builtin_amdgcn_wmma_f32_16x16x4_f32(
      /*neg_a=*/false, a, /*neg_b=*/false, bm,
      /*c_mod=*/(short)0, c, /*reuse_a=*/false, /*reuse_b=*/false);
  float s = c[0];                     // lane L holds pairsum[L % 16]
  s += __shfl_xor(s, 8, 32);
  s += __shfl_xor(s, 4, 32);
  s += __shfl_xor(s, 2, 32);
  s += __shfl_xor(s, 1, 32);
  return s;                           // wave total in every lane
}

// Pass 1: one workgroup per sample. Stream 2 x n4 float4 with non-temporal
// b128 loads (read-once data; don't pollute L2), accumulate sum of squared
// diffs in fp32, reduce wave -> block, write ||diff||^2 to ws[b].
__global__ void __launch_bounds__(TPB)
norm2_kernel(const float* __restrict__ samples,
             const float* __restrict__ truth,
             float* __restrict__ norm2,
             int n4) {
  const int b = blockIdx.x;
  const size_t base4 = (size_t)b * (size_t)n4;   // float4 index of this sample's row
  const f4* __restrict__ s4 = (const f4*)samples + base4;
  const f4* __restrict__ t4 = (const f4*)truth   + base4;

  float acc0 = 0.0f, acc1 = 0.0f, acc2 = 0.0f, acc3 = 0.0f;
#pragma unroll 4
  for (int i = threadIdx.x; i < n4; i += TPB) {
    f4 sv = __builtin_nontemporal_load(s4 + i);
    f4 tv = __builtin_nontemporal_load(t4 + i);
    f4 d  = sv - tv;
    acc0 = fmaf(d.x, d.x, acc0);
    acc1 = fmaf(d.y, d.y, acc1);
    acc2 = fmaf(d.z, d.z, acc2);
    acc3 = fmaf(d.w, d.w, acc3);
  }
  float acc = (acc0 + acc1) + (acc2 + acc3);

  // n4 (6144) is a multiple of TPB -> no divergence, EXEC all ones here.
  float wsum = wave_sum32(acc);

  __shared__ float lds[TPB / 32];
  const int wid  = threadIdx.x >> 5;
  const int lane = threadIdx.x & 31;
  if (lane == 0) lds[wid] = wsum;
  __syncthreads();
  if (threadIdx.x == 0) {
    float s = 0.0f;
#pragma unroll
    for (int w = 0; w < TPB / 32; ++w) s += lds[w];
    norm2[b] = s;
  }
}

// Pass 2: single block. per_sample = 0.5 + |d| + l2 * sqrt(norm2); mean over B.
__global__ void __launch_bounds__(TPB)
finalize_kernel(const float* __restrict__ norm2,
                const float* __restrict__ dvec,
                const int* __restrict__ iter_p,
                float* __restrict__ out,
                int B) {
  const float l2 = (15.0f / 1000.0f) * (float)(*iter_p) + 1.0f;  // ITERS = 1000
  float acc = 0.0f;
  for (int i = threadIdx.x; i < B; i += TPB) {                   // B multiple of TPB
    acc += 0.5f + fabsf(dvec[i]) + l2 * sqrtf(norm2[i]);         // LAMBDA_1 = 0.5
  }

  float wsum = wave_sum32(acc);

  __shared__ float lds[TPB / 32];
  const int wid  = threadIdx.x >> 5;
  const int lane = threadIdx.x & 31;
  if (lane == 0) lds[wid] = wsum;
  __syncthreads();
  if (threadIdx.x == 0) {
    float s = 0.0f;
#pragma unroll
    for (int w = 0; w < TPB / 32; ++w) s += lds[w];
    out[0] = s / (float)B;
  }
}

extern "C" void kernel_launch(void* const* d_in, const int* in_sizes, int n_in,
                              void* d_out, int out_size, void* d_ws, size_t ws_size,
                              hipStream_t stream) {
  const float* samples = (const float*)d_in[0];   // [B, N, C] fp32
  const float* truth   = (const float*)d_in[1];   // [B, N, C] fp32
  const float* dvec    = (const float*)d_in[2];   // [B] fp32
  const int*   iter_p  = (const int*)d_in[3];     // scalar int

  const int B  = in_sizes[2];            // 1024
  const int nc = in_sizes[0] / B;        // 24576 floats per sample
  const int n4 = nc / 4;                 // 6144 float4 per sample

  float* norm2 = (float*)d_ws;           // B floats of scratch (written every call)

  norm2_kernel<<<B, TPB, 0, stream>>>(samples, truth, norm2, n4);
  finalize_kernel<<<1, TPB, 0, stream>>>(norm2, dvec, iter_p, (float*)d_out, B);
}